// GcnForPacking_68461778698647
// MI455X (gfx1250) — compile-verified
//
#include <hip/hip_runtime.h>
#include <hip/hip_bf16.h>

typedef __attribute__((ext_vector_type(2))) float v2f;
typedef __attribute__((ext_vector_type(8))) float v8f;

#define F_IN 128
#define HID 64
#define NCLS 10

// ---------------------------------------------------------------------------
// Zero a float region (grid-stride).
// ---------------------------------------------------------------------------
__global__ void zero_f32_kernel(float* __restrict__ p, int n) {
    int i = blockIdx.x * blockDim.x + threadIdx.x;
    int stride = gridDim.x * blockDim.x;
    for (; i < n; i += stride) p[i] = 0.0f;
}

// ---------------------------------------------------------------------------
// deg[col[e]] += 1  (in-degree at targets)
// ---------------------------------------------------------------------------
__global__ void degree_kernel(const int* __restrict__ col, float* __restrict__ deg, int E) {
    int e = blockIdx.x * blockDim.x + threadIdx.x;
    if (e < E) atomicAdd(&deg[col[e]], 1.0f);
}

// ---------------------------------------------------------------------------
// deg -> dinv in place: dinv = deg > 0 ? rsqrt(deg) : 0
// ---------------------------------------------------------------------------
__global__ void dinv_kernel(float* __restrict__ deg, int N) {
    int i = blockIdx.x * blockDim.x + threadIdx.x;
    if (i < N) {
        float d = deg[i];
        deg[i] = (d > 0.0f) ? rsqrtf(fmaxf(d, 1.0f)) : 0.0f;
    }
}

// ---------------------------------------------------------------------------
// h = x @ conv_W   [N,128] @ [128,64] -> [N,64]   via V_WMMA_F32_16X16X4_F32
// Block = 128 threads = 4 waves. Block handles 16 rows; wave w handles
// columns [16w, 16w+16). K-loop: 32 WMMAs of K=4.
//
// Fragment layouts (ISA 7.12.2, 32-bit):
//   A 16x4 : lanes 0-15 -> M=lane, VGPR0=K0,VGPR1=K1 ; lanes 16-31 -> K2,K3
//   B 4x16 : lanes 0-15 -> N=lane, VGPR0=K0,VGPR1=K1 ; lanes 16-31 -> K2,K3
//   C 16x16: VGPR v, lanes 0-15 -> M=v,N=lane ; lanes 16-31 -> M=v+8
// ---------------------------------------------------------------------------
#define XS_STRIDE 132   // 128 + 4 pad: conflict-free strided row reads
#define WS_STRIDE 65    // 64 + 1 pad : conflict-free half-wave row split

__global__ __launch_bounds__(128) void gemm_xw_kernel(const float* __restrict__ x,
                                                      const float* __restrict__ W,
                                                      float* __restrict__ h, int N) {
    __shared__ float xs[16 * XS_STRIDE];
    __shared__ float ws[F_IN * WS_STRIDE];

    const int tid = threadIdx.x;
    const int rowBase = blockIdx.x * 16;

    // Stage conv_W (32 KB) into LDS, padded stride.
    for (int i = tid; i < F_IN * HID; i += 128) {
        int k = i >> 6;          // 0..127
        int c = i & 63;          // 0..63
        ws[k * WS_STRIDE + c] = W[i];
    }
    // Stage 16x128 x tile into LDS (coalesced global reads).
    for (int i = tid; i < 16 * F_IN; i += 128) {
        int r = i >> 7;          // 0..15
        int c = i & 127;         // 0..127
        int row = rowBase + r;
        xs[r * XS_STRIDE + c] = (row < N) ? x[row * F_IN + c] : 0.0f;
    }
    __syncthreads();

    const int wave = tid >> 5;           // 0..3
    const int lane = tid & 31;
    const int colBase = wave * 16;
    const int m = lane & 15;             // M for A, N for B
    const int koff = (lane < 16) ? 0 : 2;

    v8f acc = {};
    #pragma unroll 4
    for (int kk = 0; kk < F_IN; kk += 4) {
        v2f a = *(const v2f*)&xs[m * XS_STRIDE + kk + koff];   // 8B aligned
        v2f b;
        b.x = ws[(kk + koff)     * WS_STRIDE + colBase + m];
        b.y = ws[(kk + koff + 1) * WS_STRIDE + colBase + m];
        acc = __builtin_amdgcn_wmma_f32_16x16x4_f32(
            /*neg_a=*/false, a, /*neg_b=*/false, b,
            /*c_mod=*/(short)0, acc, /*reuse_a=*/false, /*reuse_b=*/false);
    }

    const int mbase = (lane < 16) ? 0 : 8;
    #pragma unroll
    for (int v = 0; v < 8; ++v) {
        int row = rowBase + mbase + v;
        if (row < N) h[row * HID + colBase + m] = acc[v];
    }
}

// ---------------------------------------------------------------------------
// Edge scatter: agg[col[e]] += h[row[e]] * (dinv[row]*dinv[col])
// One thread per (edge, float4-chunk): 16 chunks cover HID=64.
// agg footprint is 25.6 MB -> atomics resolve in the 192 MB L2.
// ---------------------------------------------------------------------------
__global__ __launch_bounds__(256) void scatter_kernel(const int* __restrict__ row,
                                                      const int* __restrict__ col,
                                                      const float* __restrict__ dinv,
                                                      const float* __restrict__ h,
                                                      float* __restrict__ agg, int E) {
    int t = blockIdx.x * blockDim.x + threadIdx.x;
    int e = t >> 4;
    if (e >= E) return;
    int c = (t & 15) << 2;                 // feature offset 0,4,...,60
    int r  = row[e];
    int cl = col[e];
    float nrm = dinv[r] * dinv[cl];
    const float4 hv = *(const float4*)&h[r * HID + c];
    float* dst = &agg[cl * HID + c];
    atomicAdd(dst + 0, hv.x * nrm);
    atomicAdd(dst + 1, hv.y * nrm);
    atomicAdd(dst + 2, hv.z * nrm);
    atomicAdd(dst + 3, hv.w * nrm);
}

// ---------------------------------------------------------------------------
// relu(agg + b) then mean-pool accumulation into per-graph sums/counts.
// One thread per (node, feature).
// ---------------------------------------------------------------------------
__global__ __launch_bounds__(256) void pool_kernel(const float* __restrict__ agg,
                                                   const float* __restrict__ conv_b,
                                                   const int* __restrict__ batch,
                                                   float* __restrict__ sums,
                                                   float* __restrict__ counts, int N) {
    int t = blockIdx.x * blockDim.x + threadIdx.x;
    int n = t >> 6;
    if (n >= N) return;
    int f = t & 63;
    float v = fmaxf(agg[n * HID + f] + conv_b[f], 0.0f);
    int g = batch[n];
    atomicAdd(&sums[g * HID + f], v);
    if (f == 0) atomicAdd(&counts[g], 1.0f);
}

// ---------------------------------------------------------------------------
// logits = (sums/counts) @ lin_W + lin_b   [G,64]@[64,10] — tiny, scalar.
// ---------------------------------------------------------------------------
__global__ void logits_kernel(const float* __restrict__ sums,
                              const float* __restrict__ counts,
                              const float* __restrict__ linW,
                              const float* __restrict__ linb,
                              float* __restrict__ out, int G) {
    int t = blockIdx.x * blockDim.x + threadIdx.x;
    if (t >= G * NCLS) return;
    int g = t / NCLS;
    int c = t - g * NCLS;
    float inv = 1.0f / fmaxf(counts[g], 1.0f);
    float acc = linb[c];
    const float* s = sums + g * HID;
    #pragma unroll
    for (int k = 0; k < HID; ++k) acc += (s[k] * inv) * linW[k * NCLS + c];
    out[t] = acc;
}

// ---------------------------------------------------------------------------
extern "C" void kernel_launch(void* const* d_in, const int* in_sizes, int n_in,
                              void* d_out, int out_size, void* d_ws, size_t ws_size,
                              hipStream_t stream) {
    const float* x      = (const float*)d_in[0];
    const int*   eidx   = (const int*)d_in[1];
    const int*   batch  = (const int*)d_in[2];
    // d_in[3] = graphs_mask (unused), d_in[4] = y (unused)
    const float* conv_W = (const float*)d_in[5];
    const float* conv_b = (const float*)d_in[6];
    const float* lin_W  = (const float*)d_in[7];
    const float* lin_b  = (const float*)d_in[8];
    float* out = (float*)d_out;

    const int N = in_sizes[0] / F_IN;      // 100000
    const int E = in_sizes[1] / 2;         // 1000000
    const int G = in_sizes[3];             // 1024

    const int* erow = eidx;                // edge_index[0]
    const int* ecol = eidx + E;            // edge_index[1]

    // Workspace layout (floats), zeroed region first, h last:
    //   deg/dinv [N] | agg [N*64] | sums [G*64] | counts [G] | h [N*64]
    float* ws     = (float*)d_ws;
    float* deg    = ws;
    float* agg    = deg + N;
    float* sums   = agg + (size_t)N * HID;
    float* counts = sums + (size_t)G * HID;
    float* h      = counts + G;

    const int zeroN = N + N * HID + G * HID + G;   // contiguous accumulators
    zero_f32_kernel<<<4096, 256, 0, stream>>>(ws, zeroN);

    degree_kernel<<<(E + 255) / 256, 256, 0, stream>>>(ecol, deg, E);
    dinv_kernel<<<(N + 255) / 256, 256, 0, stream>>>(deg, N);

    gemm_xw_kernel<<<(N + 15) / 16, 128, 0, stream>>>(x, conv_W, h, N);

    {
        long long threads = (long long)E * 16;
        int blocks = (int)((threads + 255) / 256);
        scatter_kernel<<<blocks, 256, 0, stream>>>(erow, ecol, deg, h, agg, E);
    }
    {
        long long threads = (long long)N * HID;
        int blocks = (int)((threads + 255) / 256);
        pool_kernel<<<blocks, 256, 0, stream>>>(agg, conv_b, batch, sums, counts, N);
    }
    logits_kernel<<<(G * NCLS + 255) / 256, 256, 0, stream>>>(sums, counts, lin_W, lin_b, out, G);
}